// SheafLoss_34179349741657
// MI455X (gfx1250) — compile-verified
//
#include <hip/hip_runtime.h>
#include <math.h>

typedef __attribute__((ext_vector_type(2))) float v2f;
typedef __attribute__((ext_vector_type(8))) float v8f;

#define TAU_F     1.0f
#define EPS_MASS  1e-6f
#define EPS_LOG   1e-8f
#define EPS_DEN   1e-8f

#define PP 16   // patches per (b,t)
#define DD 64   // feature dim
#define RR 32   // num centers
#define KK 64   // distribution dim
#define NPAIR 496  // RR*(RR-1)/2

// ---------------------------------------------------------------------------
// Kernel 0: precompute omega[r][s] = exp(-||c_r - c_s||^2) and cnorm[r]
// into workspace. One tiny block; results live in L2 for all main blocks.
// ---------------------------------------------------------------------------
__global__ __launch_bounds__(256) void sheaf_init_kernel(
    const float* __restrict__ gC, float* __restrict__ ws) {
  const int tid = threadIdx.x;
  for (int e = tid; e < RR * RR; e += 256) {
    const int i = e >> 5, j = e & 31;
    float d2 = 0.f;
    #pragma unroll 8
    for (int d = 0; d < DD; ++d) {
      const float df = gC[i * DD + d] - gC[j * DD + d];
      d2 += df * df;
    }
    ws[e] = __expf(-d2 / TAU_F);
  }
  if (tid < RR) {
    float n = 0.f;
    #pragma unroll 8
    for (int d = 0; d < DD; ++d) { const float c = gC[tid * DD + d]; n += c * c; }
    ws[RR * RR + tid] = n;
  }
}

// ---------------------------------------------------------------------------
// Kernel 1: one workgroup per (b,t).  WMMA for both small GEMMs, VALU/TRANS
// for softmax + JS pair loop.  Writes one partial sum per block.
// ---------------------------------------------------------------------------
__global__ __launch_bounds__(256) void sheaf_main_kernel(
    const float* __restrict__ gM, const float* __restrict__ gW,
    const float* __restrict__ gP, const float* __restrict__ gC,
    const float* __restrict__ ws, float* __restrict__ partial) {
  __shared__ float sM[PP][DD + 1];     // m tile          (padded vs 64 banks)
  __shared__ float sP[PP][DD + 1];     // p tile
  __shared__ float sC[RR][DD + 1];     // centers
  __shared__ float sPbar[RR][KK + 1];  // p_bar
  __shared__ float sS[PP][RR + 1];     // logits -> start_mass
  __shared__ float sOm[RR][RR + 1];    // omega
  __shared__ float sCn[RR];            // ||c_r||^2
  __shared__ float sTm[RR];            // total_mass_r
  __shared__ float sHr[RR];            // sum_k p_bar log(p_bar+eps)
  __shared__ float sW[PP];
  __shared__ float sRed[256];

  const int tid  = threadIdx.x;
  const int bid  = blockIdx.x;
  const int lane = tid & 31;
  const int wv   = tid >> 5;

  // ---- cooperative loads -------------------------------------------------
  const size_t base = (size_t)bid * PP * DD;
  for (int e = tid; e < PP * DD; e += 256) {
    sM[e >> 6][e & 63] = gM[base + e];
    sP[e >> 6][e & 63] = gP[base + e];
  }
  for (int e = tid; e < RR * DD; e += 256) sC[e >> 6][e & 63] = gC[e];
  for (int e = tid; e < RR * RR; e += 256) sOm[e >> 5][e & 31] = ws[e];
  if (tid < RR) { sCn[tid] = ws[RR * RR + tid]; sHr[tid] = 0.f; }
  if (tid < PP) sW[tid] = gW[(size_t)bid * PP + tid];
  __syncthreads();

  // ---- stage B: logits[p][r] = 2*(m . c_r) - ||c_r||^2 via WMMA ----------
  // G (16x32) = m (16x64) @ c^T (64x32): two 16x16 N-tiles, 16 K-chunks of 4.
  // A 16x4 f32 layout: lane L (0-15) holds (K=0,K=1); lane L+16 holds (K=2,K=3).
  // B 4x16  f32 layout: lane n (0-15) holds rows K=0/K=1; lane n+16 rows K=2/K=3.
  if (wv < 2) {
    const int r0   = wv * 16;
    const int n    = lane & 15;
    const int koff = (lane >> 4) * 2;
    v8f acc = {0.f, 0.f, 0.f, 0.f, 0.f, 0.f, 0.f, 0.f};
    #pragma unroll
    for (int d0 = 0; d0 < DD; d0 += 4) {
      v2f a, b;
      a.x = sM[n][d0 + koff];      a.y = sM[n][d0 + koff + 1];
      b.x = sC[r0 + n][d0 + koff]; b.y = sC[r0 + n][d0 + koff + 1];
      acc = __builtin_amdgcn_wmma_f32_16x16x4_f32(
          false, a, false, b, (short)0, acc, false, false);
    }
    // C/D layout: lane n, vgpr v -> (M = v + 8*(lane>=16), N = n)
    const int mb = (lane >> 4) * 8;
    #pragma unroll
    for (int v = 0; v < 8; ++v) {
      const int r = r0 + n;
      sS[mb + v][r] = 2.f * acc[v] - sCn[r];
    }
  }
  __syncthreads();

  // ---- stage C: softmax over r, scale by w[p] ----------------------------
  if (tid < PP) {
    float mx = -3.4e38f;
    for (int r = 0; r < RR; ++r) mx = fmaxf(mx, sS[tid][r]);
    float sum = 0.f;
    for (int r = 0; r < RR; ++r) {
      const float e = __expf(sS[tid][r] - mx);
      sS[tid][r] = e;
      sum += e;
    }
    const float scl = sW[tid] / sum;
    for (int r = 0; r < RR; ++r) sS[tid][r] *= scl;
  }
  __syncthreads();
  if (tid < RR) {
    float t = 0.f;
    #pragma unroll
    for (int pp = 0; pp < PP; ++pp) t += sS[pp][tid];
    sTm[tid] = t;
  }
  __syncthreads();

  // ---- stage D: p_bar (32x64) = S^T (32x16) @ p (16x64) via WMMA ---------
  // 8 waves, each owns one (Mtile,Ntile) 16x16 output; 4 K-chunks of 4.
  {
    const int r0   = (wv >> 2) * 16;
    const int n0   = (wv & 3) * 16;
    const int n    = lane & 15;
    const int koff = (lane >> 4) * 2;
    v8f acc = {0.f, 0.f, 0.f, 0.f, 0.f, 0.f, 0.f, 0.f};
    #pragma unroll
    for (int p0 = 0; p0 < PP; p0 += 4) {
      v2f a, b;
      a.x = sS[p0 + koff][r0 + n];     a.y = sS[p0 + koff + 1][r0 + n];   // A[r][p]=S[p][r]
      b.x = sP[p0 + koff][n0 + n];     b.y = sP[p0 + koff + 1][n0 + n];
      acc = __builtin_amdgcn_wmma_f32_16x16x4_f32(
          false, a, false, b, (short)0, acc, false, false);
    }
    const int mb = (lane >> 4) * 8;
    #pragma unroll
    for (int v = 0; v < 8; ++v) {
      const int r = r0 + mb + v;
      sPbar[r][n0 + n] = acc[v] / (sTm[r] + EPS_MASS);
    }
  }
  __syncthreads();

  // ---- stage E: H_r = sum_k p_bar log(p_bar + eps) -----------------------
  {
    const int r  = tid & 31;
    const int k0 = (tid >> 5) * 8;
    float h = 0.f;
    #pragma unroll
    for (int k = k0; k < k0 + 8; ++k) {
      const float v = sPbar[r][k];
      h += v * __logf(v + EPS_LOG);
    }
    atomicAdd(&sHr[r], h);
  }
  __syncthreads();

  // ---- stage F: 496 pairs, js = 0.5(H_r+H_s) - sum_k m log(m+eps) --------
  float local = 0.f;
  for (int pi = tid; pi < NPAIR; pi += 256) {
    int idx = pi, r = 0, cnt = RR - 1;
    while (idx >= cnt) { idx -= cnt; --cnt; ++r; }
    const int s = r + 1 + idx;
    float cross = 0.f;
    #pragma unroll 8
    for (int k = 0; k < KK; ++k) {
      const float mm = 0.5f * (sPbar[r][k] + sPbar[s][k]);
      cross += mm * __logf(mm + EPS_LOG);
    }
    const float js = 0.5f * (sHr[r] + sHr[s]) - cross;
    local += sOm[r][s] * js;
  }

  // ---- block reduce -> one partial per block -----------------------------
  sRed[tid] = local;
  __syncthreads();
  #pragma unroll
  for (int st = 128; st > 0; st >>= 1) {
    if (tid < st) sRed[tid] += sRed[tid + st];
    __syncthreads();
  }
  if (tid == 0) partial[bid] = sRed[0];
}

// ---------------------------------------------------------------------------
// Kernel 2: deterministic final reduction (no global float atomics).
// ---------------------------------------------------------------------------
__global__ __launch_bounds__(256) void sheaf_reduce_kernel(
    const float* __restrict__ partial, int n, float* __restrict__ out) {
  __shared__ float sRed[256];
  float s = 0.f;
  for (int i = threadIdx.x; i < n; i += 256) s += partial[i];
  sRed[threadIdx.x] = s;
  __syncthreads();
  #pragma unroll
  for (int st = 128; st > 0; st >>= 1) {
    if (threadIdx.x < st) sRed[threadIdx.x] += sRed[threadIdx.x + st];
    __syncthreads();
  }
  if (threadIdx.x == 0) out[0] = sRed[0] / ((float)NPAIR + EPS_DEN);
}

// ---------------------------------------------------------------------------
extern "C" void kernel_launch(void* const* d_in, const int* in_sizes, int n_in,
                              void* d_out, int out_size, void* d_ws, size_t ws_size,
                              hipStream_t stream) {
  const float* gM = (const float*)d_in[0];   // (B,T,P,D)
  const float* gW = (const float*)d_in[1];   // (B,T,P)
  const float* gP = (const float*)d_in[2];   // (B,T,P,K)
  const float* gC = (const float*)d_in[3];   // (R,D)
  float* ws  = (float*)d_ws;
  float* out = (float*)d_out;

  const int BT = in_sizes[1] / PP;           // B*T blocks
  float* partial = ws + RR * RR + RR;        // after omega + cnorm

  sheaf_init_kernel<<<1, 256, 0, stream>>>(gC, ws);
  sheaf_main_kernel<<<BT, 256, 0, stream>>>(gM, gW, gP, gC, ws, partial);
  sheaf_reduce_kernel<<<1, 256, 0, stream>>>(partial, BT, out);
}